// DensityLoss_58686433133076
// MI455X (gfx1250) — compile-verified
//
#include <hip/hip_runtime.h>
#include <hip/hip_bf16.h>

typedef float v2f __attribute__((ext_vector_type(2)));
typedef float v8f __attribute__((ext_vector_type(8)));

#define BB 8
#define HH 512
#define WW 512
#define NPTS 32
#define HW (HH * WW)
#define EPSV 1e-8f
// -1/(2*0.1^2) * log2(e)
#define CEXP2 (-72.13475204444817f)

// meta layout (floats, after targ[B*H*W]):
//  [0..7]   S_p per batch
//  [8..15]  S_t per batch
//  [16..23] pred argmax idx (as int bits)
//  [24..31] targ argmax idx (as int bits)
//  [32..39] kl partial per batch
//  [40..55] emd_h partials (16 blocks)
//  [56..567] emd_w partials (512 blocks)

// ds_swizzle group-of-32: and_mask=0x1f, or_mask=0, xor_mask=0x10 (SWAPX16)
__device__ __forceinline__ float xor16_add(float s) {
  int t = __builtin_amdgcn_ds_swizzle(__float_as_int(s), 0x401f);
  return s + __int_as_float(t);
}

// ---------------------------------------------------------------------------
// Kernel 1: KDE target density via V_WMMA_F32_16X16X4_F32.
// CEXP2*dist^2(p,g) folded into the matrix product:
//   A[m] = CEXP2 * [-2px, -2py, px^2+py^2, 1]   (rows = control points,
//                                                loop-invariant per block)
//   B[:,n] = [gx, gy, 1, gx^2+gy^2]             (cols = grid cells)
// D layout: lane = grid cell, VGPR r / half = point -> sum over 32 points is a
// per-lane tree over 16 registers + ONE ds_swizzle(SWAPX16).
// Tiles processed in pairs so the store is full-EXEC and 128B-coalesced.
// One block per (b, h); 8 waves x 2 tile-pairs x 32 cells = 512-cell row.
// ---------------------------------------------------------------------------
__global__ __launch_bounds__(256) void kde_wmma_kernel(
    const float* __restrict__ pts, float* __restrict__ targ) {
  const int b = blockIdx.x / HH;
  const int h = blockIdx.x % HH;
  const int lane = threadIdx.x & 31;
  const int wave = threadIdx.x >> 5;
  const int nlo = lane & 15;
  const int half = (lane >> 4) & 1;
  const float inv = 1.0f / (float)(WW - 1);
  const float gy = (float)h * inv;
  const float gy2 = gy * gy;

  // A operand, chunk0 = points 0..15, chunk1 = points 16..31 (CEXP2-scaled).
  // lanes 0-15: row m=nlo, K=0,1 = CEXP2*(-2px, -2py)
  // lanes 16-31: row m=nlo, K=2,3 = CEXP2*(px^2+py^2, 1)
  const float* P = pts + b * (NPTS * 2);
  const float px0 = P[2 * nlo], py0 = P[2 * nlo + 1];
  const float px1 = P[2 * (nlo + 16)], py1 = P[2 * (nlo + 16) + 1];
  v2f am0, am1;
  am0.x = half ? (CEXP2 * (px0 * px0 + py0 * py0)) : (-2.0f * CEXP2 * px0);
  am0.y = half ? CEXP2 : (-2.0f * CEXP2 * py0);
  am1.x = half ? (CEXP2 * (px1 * px1 + py1 * py1)) : (-2.0f * CEXP2 * px1);
  am1.y = half ? CEXP2 : (-2.0f * CEXP2 * py1);

  float* rowout = targ + ((size_t)b * HH + h) * WW;

  for (int i = 0; i < 2; ++i) {
    const int w0A = (wave * 2 + i) * 32;  // tile pair covers 32 cells
    float s[2];
#pragma unroll
    for (int t = 0; t < 2; ++t) {
      const int w0 = w0A + t * 16;
      const float gx = (float)(w0 + nlo) * inv;
      // B operand: col n=nlo. lanes 0-15: K=0,1 = (gx, gy);
      // lanes 16-31: K=2,3 = (1, gx^2+gy^2)
      v2f bm;
      bm.x = half ? 1.0f : gx;
      bm.y = half ? (gx * gx + gy2) : gy;

      v8f c0 = {};
      v8f c1 = {};
      c0 = __builtin_amdgcn_wmma_f32_16x16x4_f32(false, am0, false, bm, (short)0, c0, false, false);
      c1 = __builtin_amdgcn_wmma_f32_16x16x4_f32(false, am1, false, bm, (short)0, c1, false, false);

      float e[8];
#pragma unroll
      for (int r = 0; r < 8; ++r)
        e[r] = __builtin_amdgcn_exp2f(c0[r]) + __builtin_amdgcn_exp2f(c1[r]);

      // per-lane tree (16 of 32 points), then merge complementary halves
      float p = ((e[0] + e[1]) + (e[2] + e[3])) + ((e[4] + e[5]) + (e[6] + e[7]));
      s[t] = xor16_add(p);  // all 32 lanes now hold the full sum
    }
    // full-wave coalesced store: lanes 0-15 -> tile 0, lanes 16-31 -> tile 1
    rowout[w0A + half * 16 + nlo] = half ? s[1] : s[0];
  }
}

// ---------------------------------------------------------------------------
// Kernel 2: per-batch sums (normalization) + first-occurrence argmax (peaks)
// ---------------------------------------------------------------------------
__global__ __launch_bounds__(256) void batch_reduce_kernel(
    const float* __restrict__ pred, const float* __restrict__ targ,
    const float* __restrict__ mask, float* __restrict__ meta) {
  const int b = blockIdx.x;
  const int tid = threadIdx.x;
  const float mk = mask[b];
  const float* P = pred + (size_t)b * HW;
  const float* T = targ + (size_t)b * HW;
  float sp = 0.f, st = 0.f, pv = -1e30f, tv = -1e30f;
  int pi = 0, ti = 0;
  for (int i = tid; i < HW; i += 256) {
    float p = P[i] * mk;
    float t = T[i] * mk;
    sp += p;
    st += t;
    if (p > pv) { pv = p; pi = i; }
    if (t > tv) { tv = t; ti = i; }
  }
  __shared__ float s_sp[256], s_st[256], s_pv[256], s_tv[256];
  __shared__ int s_pi[256], s_ti[256];
  s_sp[tid] = sp; s_st[tid] = st;
  s_pv[tid] = pv; s_pi[tid] = pi;
  s_tv[tid] = tv; s_ti[tid] = ti;
  __syncthreads();
  for (int s = 128; s > 0; s >>= 1) {
    if (tid < s) {
      s_sp[tid] += s_sp[tid + s];
      s_st[tid] += s_st[tid + s];
      float ov = s_pv[tid + s]; int oi = s_pi[tid + s];
      if (ov > s_pv[tid] || (ov == s_pv[tid] && oi < s_pi[tid])) { s_pv[tid] = ov; s_pi[tid] = oi; }
      ov = s_tv[tid + s]; oi = s_ti[tid + s];
      if (ov > s_tv[tid] || (ov == s_tv[tid] && oi < s_ti[tid])) { s_tv[tid] = ov; s_ti[tid] = oi; }
    }
    __syncthreads();
  }
  if (tid == 0) {
    meta[b] = s_sp[0] + (float)HW * EPSV;
    meta[8 + b] = s_st[0] + (float)HW * EPSV;
    ((int*)meta)[16 + b] = s_pi[0];
    ((int*)meta)[24 + b] = s_ti[0];
  }
}

// ---------------------------------------------------------------------------
// Kernel 3a: KL divergence partial per batch
// ---------------------------------------------------------------------------
__global__ __launch_bounds__(256) void kl_kernel(
    const float* __restrict__ pred, const float* __restrict__ targ,
    const float* __restrict__ mask, float* __restrict__ meta) {
  const int b = blockIdx.x;
  const int tid = threadIdx.x;
  const float mk = mask[b];
  const float invSp = 1.0f / meta[b];
  const float invSt = 1.0f / meta[8 + b];
  const float* P = pred + (size_t)b * HW;
  const float* T = targ + (size_t)b * HW;
  float acc = 0.f;
  for (int i = tid; i < HW; i += 256) {
    float pn = (P[i] * mk + EPSV) * invSp;
    float tn = (T[i] * mk + EPSV) * invSt;
    acc += tn * (__logf(tn) - __logf(pn + EPSV));
  }
  __shared__ float red[256];
  red[tid] = acc;
  __syncthreads();
  for (int s = 128; s > 0; s >>= 1) {
    if (tid < s) red[tid] += red[tid + s];
    __syncthreads();
  }
  if (tid == 0) meta[32 + b] = red[0];
}

// ---------------------------------------------------------------------------
// Kernel 3b: EMD along H (cumsum over rows). One thread per (b, w) column;
// consecutive threads = consecutive w -> coalesced row reads at each h step.
// ---------------------------------------------------------------------------
__global__ __launch_bounds__(256) void emd_h_kernel(
    const float* __restrict__ pred, const float* __restrict__ targ,
    const float* __restrict__ mask, float* __restrict__ meta) {
  const int col = blockIdx.x * 256 + threadIdx.x;  // 0..4095
  const int b = col >> 9;
  const int w = col & (WW - 1);
  const float mk = mask[b];
  const float invSp = 1.0f / meta[b];
  const float invSt = 1.0f / meta[8 + b];
  float run = 0.f, acc = 0.f;
  for (int h = 0; h < HH; ++h) {
    size_t idx = ((size_t)b * HH + h) * WW + w;
    float d = (pred[idx] * mk + EPSV) * invSp - (targ[idx] * mk + EPSV) * invSt;
    run += d;
    acc += fabsf(run);
  }
  __shared__ float red[256];
  red[threadIdx.x] = acc;
  __syncthreads();
  for (int s = 128; s > 0; s >>= 1) {
    if (threadIdx.x < s) red[threadIdx.x] += red[threadIdx.x + s];
    __syncthreads();
  }
  if (threadIdx.x == 0) meta[40 + blockIdx.x] = red[0];
}

// ---------------------------------------------------------------------------
// Kernel 3c: EMD along W (cumsum over columns). One wave32 per (b, h) row,
// shuffle-based inclusive scan over 16 chunks of 32 with running carry.
// ---------------------------------------------------------------------------
__global__ __launch_bounds__(256) void emd_w_kernel(
    const float* __restrict__ pred, const float* __restrict__ targ,
    const float* __restrict__ mask, float* __restrict__ meta) {
  const int lane = threadIdx.x & 31;
  const int wave = threadIdx.x >> 5;
  const int row = blockIdx.x * 8 + wave;  // 0..4095
  const int b = row >> 9;
  const int h = row & (HH - 1);
  const float mk = mask[b];
  const float invSp = 1.0f / meta[b];
  const float invSt = 1.0f / meta[8 + b];
  float carry = 0.f, acc = 0.f;
  const size_t base = ((size_t)b * HH + h) * WW;
  for (int chunk = 0; chunk < 16; ++chunk) {
    size_t idx = base + chunk * 32 + lane;
    float d = (pred[idx] * mk + EPSV) * invSp - (targ[idx] * mk + EPSV) * invSt;
#pragma unroll
    for (int off = 1; off < 32; off <<= 1) {
      float n = __shfl_up(d, off, 32);
      if (lane >= off) d += n;
    }
    d += carry;
    acc += fabsf(d);
    carry = __shfl(d, 31, 32);
  }
#pragma unroll
  for (int m = 16; m > 0; m >>= 1) acc += __shfl_xor(acc, m, 32);
  __shared__ float red[8];
  if (lane == 0) red[wave] = acc;
  __syncthreads();
  if (threadIdx.x == 0) {
    float s = 0.f;
    for (int i = 0; i < 8; ++i) s += red[i];
    meta[56 + blockIdx.x] = s;
  }
}

// ---------------------------------------------------------------------------
// Kernel 4: finalize -> d_out[0]=loss, [1]=emd, [2]=peak_accuracy
// ---------------------------------------------------------------------------
__global__ void finalize_kernel(const float* __restrict__ meta,
                                float* __restrict__ out) {
  float kl = 0.f;
  for (int b = 0; b < BB; ++b) kl += meta[32 + b];
  float eh = 0.f;
  for (int i = 0; i < 16; ++i) eh += meta[40 + i];
  float ew = 0.f;
  for (int i = 0; i < 512; ++i) ew += meta[56 + i];
  const float invHW = 1.0f / (float)((size_t)BB * HW);
  float emd = 0.5f * (eh * invHW + ew * invHW);

  float cnt = 0.f;
  for (int b = 0; b < BB; ++b) {
    int pi = ((const int*)meta)[16 + b];
    int ti = ((const int*)meta)[24 + b];
    float ppx = (float)(pi & (WW - 1)) / (float)(WW - 1);
    float ppy = (float)(pi >> 9) / (float)(HH - 1);
    float tpx = (float)(ti & (WW - 1)) / (float)(WW - 1);
    float tpy = (float)(ti >> 9) / (float)(HH - 1);
    float dx = ppx - tpx, dy = ppy - tpy;
    if (sqrtf(dx * dx + dy * dy) < 0.1f) cnt += 1.0f;
  }
  out[0] = kl / (float)BB;
  out[1] = emd;
  out[2] = cnt / (float)BB;
}

extern "C" void kernel_launch(void* const* d_in, const int* in_sizes, int n_in,
                              void* d_out, int out_size, void* d_ws, size_t ws_size,
                              hipStream_t stream) {
  const float* pred = (const float*)d_in[0];  // [B,H,W]
  const float* pts = (const float*)d_in[1];   // [B,N,2]
  const float* mask = (const float*)d_in[2];  // [B]
  float* out = (float*)d_out;

  float* targ = (float*)d_ws;                 // B*H*W floats
  float* meta = targ + (size_t)BB * HW;       // ~600 floats

  kde_wmma_kernel<<<BB * HH, 256, 0, stream>>>(pts, targ);
  batch_reduce_kernel<<<BB, 256, 0, stream>>>(pred, targ, mask, meta);
  kl_kernel<<<BB, 256, 0, stream>>>(pred, targ, mask, meta);
  emd_h_kernel<<<16, 256, 0, stream>>>(pred, targ, mask, meta);
  emd_w_kernel<<<512, 256, 0, stream>>>(pred, targ, mask, meta);
  finalize_kernel<<<1, 1, 0, stream>>>(meta, out);
}